// Differentiable_Voxelizer_3985729651031
// MI455X (gfx1250) — compile-verified
//
#include <hip/hip_runtime.h>
#include <math.h>

typedef float v2f __attribute__((ext_vector_type(2)));
typedef float v8f __attribute__((ext_vector_type(8)));

#define EPSF 1e-8f
#define INV4PI 0.07957747154594767f

// ---------------------------------------------------------------------------
// Zero workspace (dual_v, vn accumulators, packed A tables) — ws is poisoned.
// ---------------------------------------------------------------------------
__global__ void wf_zero_kernel(float* __restrict__ p, int n) {
  int i = blockIdx.x * blockDim.x + threadIdx.x;
  if (i < n) p[i] = 0.0f;
}

// ---------------------------------------------------------------------------
// Phase 1a: per-face dual-area weights + face normals, scattered to vertices.
// ---------------------------------------------------------------------------
__device__ __forceinline__ float corner_angle(float ux, float uy, float uz,
                                              float vx, float vy, float vz) {
  float d  = ux * vx + uy * vy + uz * vz;
  float nu = __builtin_amdgcn_sqrtf(ux * ux + uy * uy + uz * uz);
  float nv = __builtin_amdgcn_sqrtf(vx * vx + vy * vy + vz * vz);
  float c  = -d / (EPSF + nu * nv);
  c = fminf(fmaxf(c, -1.0f), 1.0f);
  return acosf(c);
}

__global__ void wf_face_kernel(const float* __restrict__ verts,
                               const int* __restrict__ faces,
                               float* __restrict__ dualv,
                               float* __restrict__ vn, int F) {
  int f = blockIdx.x * blockDim.x + threadIdx.x;
  if (f >= F) return;
  int i0 = faces[3 * f + 0], i1 = faces[3 * f + 1], i2 = faces[3 * f + 2];
  float p0x = verts[3 * i0], p0y = verts[3 * i0 + 1], p0z = verts[3 * i0 + 2];
  float p1x = verts[3 * i1], p1y = verts[3 * i1 + 1], p1z = verts[3 * i1 + 2];
  float p2x = verts[3 * i2], p2y = verts[3 * i2 + 1], p2z = verts[3 * i2 + 2];

  float ax = p1x - p0x, ay = p1y - p0y, az = p1z - p0z;   // A = fv1-fv0
  float bx = p2x - p1x, by = p2y - p1y, bz = p2z - p1z;   // B = fv2-fv1
  float cx = p0x - p2x, cy = p0y - p2y, cz = p0z - p2z;   // C = fv0-fv2

  float a0 = corner_angle(cx, cy, cz, ax, ay, az);
  float a1 = corner_angle(ax, ay, az, bx, by, bz);
  float a2 = corner_angle(bx, by, bz, cx, cy, cz);

  float s0 = sinf(2.0f * a0), s1 = sinf(2.0f * a1), s2 = sinf(2.0f * a2);
  float inv = 1.0f / (s0 + s1 + s2 + EPSF);
  float w0 = s0 * inv, w1 = s1 * inv, w2 = s2 * inv;
  // w' = (w[[2,0,1]] + w[[1,2,0]]) / 2
  float wp0 = 0.5f * (w2 + w1);
  float wp1 = 0.5f * (w0 + w2);
  float wp2 = 0.5f * (w1 + w0);

  // face normal fn = cross(A, B); area = 0.5 |fn|
  float fnx = ay * bz - az * by;
  float fny = az * bx - ax * bz;
  float fnz = ax * by - ay * bx;
  float area = 0.5f * __builtin_amdgcn_sqrtf(fnx * fnx + fny * fny + fnz * fnz);

  atomicAdd(&dualv[i0], wp0 * area);
  atomicAdd(&dualv[i1], wp1 * area);
  atomicAdd(&dualv[i2], wp2 * area);
  atomicAdd(&vn[3 * i0 + 0], fnx); atomicAdd(&vn[3 * i0 + 1], fny); atomicAdd(&vn[3 * i0 + 2], fnz);
  atomicAdd(&vn[3 * i1 + 0], fnx); atomicAdd(&vn[3 * i1 + 1], fny); atomicAdd(&vn[3 * i1 + 2], fnz);
  atomicAdd(&vn[3 * i2 + 0], fnx); atomicAdd(&vn[3 * i2 + 1], fny); atomicAdd(&vn[3 * i2 + 2], fnz);
}

// ---------------------------------------------------------------------------
// Phase 1b: per-vertex areaic normal, then pack the two WMMA A-matrices in the
// exact 16x4 f32 A-fragment lane layout:
//   lane m (0..15):  VGPR0 = A[m][K=0], VGPR1 = A[m][K=1]
//   lane m+16:       VGPR0 = A[m][K=2], VGPR1 = A[m][K=3]
// Stored as float2 per lane: A[(tile*32 + lane)] = {k_even, k_odd}.
//   A1 row v = ( vx, vy, vz, |v|^2 )          -> r^2 GEMM
//   A2 row v = ( -nax, -nay, -naz, v.na )     -> numerator GEMM
// ---------------------------------------------------------------------------
__global__ void wf_vertex_kernel(const float* __restrict__ verts,
                                 const float* __restrict__ dualv,
                                 const float* __restrict__ vn,
                                 float2* __restrict__ A1,
                                 float2* __restrict__ A2, int V) {
  int v = blockIdx.x * blockDim.x + threadIdx.x;
  if (v >= V) return;
  float nx = vn[3 * v], ny = vn[3 * v + 1], nz = vn[3 * v + 2];
  float nrm = __builtin_amdgcn_sqrtf(nx * nx + ny * ny + nz * nz);
  float s = dualv[v] / (nrm + EPSF);
  float nax = nx * s, nay = ny * s, naz = nz * s;

  float x = verts[3 * v], y = verts[3 * v + 1], z = verts[3 * v + 2];
  float sq = x * x + y * y + z * z;
  float c  = x * nax + y * nay + z * naz;

  int t = v >> 4, m = v & 15;
  A1[t * 32 + m]      = make_float2(x, y);
  A1[t * 32 + m + 16] = make_float2(z, sq);
  A2[t * 32 + m]      = make_float2(-nax, -nay);
  A2[t * 32 + m + 16] = make_float2(-naz, c);
}

// ---------------------------------------------------------------------------
// Phase 2: one wave per 16-point tile. For each 16-vertex tile:
//   r2  = WMMA(A1, B1, C=|p|^2)     (B1 col p = (-2px,-2py,-2pz, 1));
//         |p|^2 depends only on the column N = lane, so broadcasting it into
//         all 8 C VGPRs folds the "+psq" into the matrix pipe for free.
//   num = WMMA(A2, B2, 0)           (B2 col p = (  px,  py,  pz, 1))
//   acc8 += num / (|r2|^1.5 + eps)  (per-pair sqrt + rcp epilogue; fabs guards
//         cancellation residue and folds into src modifiers for free)
// Accumulator layout: lane = point column N, 8 VGPRs = vertex rows M
// -> reduce over vertices = 8 independent in-lane FMAs + one xor-16 shuffle.
// ---------------------------------------------------------------------------
__global__ __launch_bounds__(256) void wf_wind_kernel(
    const float* __restrict__ points,
    const float2* __restrict__ A1,
    const float2* __restrict__ A2,
    float* __restrict__ out, int nvt, int npt) {
  int lane = threadIdx.x & 31;
  int wave = (blockIdx.x * blockDim.x + threadIdx.x) >> 5;
  if (wave >= npt) return;

  int pidx = wave * 16 + (lane & 15);
  float px = points[3 * pidx], py = points[3 * pidx + 1], pz = points[3 * pidx + 2];
  float psq = px * px + py * py + pz * pz;
  bool hi = lane >= 16;

  // B fragments (4x16, K x N): lane n: VGPR0=B[0][n], VGPR1=B[1][n];
  //                            lane n+16: VGPR0=B[2][n], VGPR1=B[3][n].
  v2f b1, b2;
  b1.x = hi ? (-2.0f * pz) : (-2.0f * px);
  b1.y = hi ? 1.0f         : (-2.0f * py);
  b2.x = hi ? pz           : px;
  b2.y = hi ? 1.0f         : py;

  // C matrix for the r^2 WMMA: every row gets this lane's |p|^2.
  v8f cpsq;
#pragma unroll
  for (int i = 0; i < 8; ++i) cpsq[i] = psq;

  v8f acc8 = {};
#pragma unroll 2
  for (int t = 0; t < nvt; ++t) {
    if (t + 8 < nvt) {
      __builtin_prefetch(&A1[(t + 8) * 32 + lane], 0, 0);
      __builtin_prefetch(&A2[(t + 8) * 32 + lane], 0, 0);
    }
    float2 f1 = A1[t * 32 + lane];
    float2 f2 = A2[t * 32 + lane];
    v2f a1; a1.x = f1.x; a1.y = f1.y;
    v2f a2; a2.x = f2.x; a2.y = f2.y;

    v8f zc = {};
    v8f r2t = __builtin_amdgcn_wmma_f32_16x16x4_f32(
        false, a1, false, b1, (short)0, cpsq, false, false);
    v8f nmt = __builtin_amdgcn_wmma_f32_16x16x4_f32(
        false, a2, false, b2, (short)0, zc, false, false);

#pragma unroll
    for (int i = 0; i < 8; ++i) {
      float r2 = __builtin_fabsf(r2t[i]);       // folds into src modifiers
      float r  = __builtin_amdgcn_sqrtf(r2);
      float den = fmaf(r2, r, EPSF);            // r^3 + eps
      acc8[i] = fmaf(nmt[i], __builtin_amdgcn_rcpf(den), acc8[i]);
    }
  }

  // reduce 8 row-accumulators, then combine row halves
  // (M=0..7 in lanes 0-15, M=8..15 in lanes 16-31)
  float acc = ((acc8[0] + acc8[1]) + (acc8[2] + acc8[3])) +
              ((acc8[4] + acc8[5]) + (acc8[6] + acc8[7]));
  acc += __shfl_xor(acc, 16, 32);
  if (lane < 16) out[wave * 16 + lane] = acc * INV4PI;
}

// ---------------------------------------------------------------------------
extern "C" void kernel_launch(void* const* d_in, const int* in_sizes, int n_in,
                              void* d_out, int out_size, void* d_ws, size_t ws_size,
                              hipStream_t stream) {
  const float* verts  = (const float*)d_in[0];
  const float* points = (const float*)d_in[1];
  const int*   faces  = (const int*)d_in[2];
  int V = in_sizes[0] / 3;
  int P = in_sizes[1] / 3;
  int F = in_sizes[2] / 3;
  int ntiles = (V + 15) / 16;

  // Workspace layout (contiguous floats): dual_v | vn | A1 | A2
  float*  dualv = (float*)d_ws;
  float*  vn    = dualv + V;
  float2* A1    = (float2*)(vn + 3 * V);
  float2* A2    = A1 + ntiles * 32;

  int nzero = 4 * V + 4 * ntiles * 32;   // floats: dualv + vn + A1 + A2
  wf_zero_kernel<<<(nzero + 255) / 256, 256, 0, stream>>>(dualv, nzero);
  wf_face_kernel<<<(F + 255) / 256, 256, 0, stream>>>(verts, faces, dualv, vn, F);
  wf_vertex_kernel<<<(V + 255) / 256, 256, 0, stream>>>(verts, dualv, vn, A1, A2, V);

  int npt = P / 16;                       // point tiles (one wave each)
  int blocks = (npt + 7) / 8;             // 8 waves (256 threads) per block
  wf_wind_kernel<<<blocks, 256, 0, stream>>>(points, A1, A2, (float*)d_out,
                                             ntiles, npt);
}